// NMSLayer_14577119002957
// MI455X (gfx1250) — compile-verified
//
#include <hip/hip_runtime.h>
#include <stdint.h>

// Problem constants (match reference)
#define BATCH     32
#define NBOX      16384
#define NTHREADS  1024              // 32 wave32 waves per WGP
#define KPT       (NBOX / NTHREADS) // 16 boxes per thread, held in registers
#define MAXOUT    100
#define IOU_THR   0.6f
#define SCORE_THR 0.5f

// Flat float offsets in d_out: [nb | ns | classes | valid]
#define NS_OFF (BATCH * MAXOUT * 4)        // 12800
#define CL_OFF (NS_OFF + BATCH * MAXOUT)   // 16000
#define VD_OFF (CL_OFF + BATCH * MAXOUT)   // 19200

__launch_bounds__(NTHREADS, 1)
__global__ void nms_gfx1250_kernel(const float* __restrict__ in,
                                   float* __restrict__ out) {
  const int tid  = threadIdx.x;
  const int b    = blockIdx.x;
  const int lane = tid & 31;
  const int wid  = tid >> 5;

  __shared__ float s_redv[32];
  __shared__ int   s_redi[32];
  __shared__ float s_box[4];
  __shared__ float s_bestv;
  __shared__ int   s_besti;
  // Output staging: [0,400) boxes, [400,500) scores, [500,600) classes, [600] valid
  __shared__ float s_stage[601];

  for (int i = tid; i < 601; i += NTHREADS) s_stage[i] = 0.0f;

  // ---- Load + decode: each thread owns 16 strided boxes in registers ----
  float bx1[KPT], by1[KPT], bx2[KPT], by2[KPT], barea[KPT], sc[KPT];
  const float* inb = in + (size_t)b * NBOX * 5;
#pragma unroll
  for (int j = 0; j < KPT; ++j) {
    const int e = j * NTHREADS + tid;             // strided: coalesced 20B records
    const float* r = inb + (size_t)e * 5;
    const float cx = r[0], cy = r[1], w = r[2], h = r[3], obj = r[4];
    const float hw = 0.5f * w, hh = 0.5f * h;
    bx1[j] = cx - hw;  by1[j] = cy - hh;
    bx2[j] = cx + hw;  by2[j] = cy + hh;
    barea[j] = w * h;                             // (x2-x1)*(y2-y1) exactly
    sc[j] = (obj >= SCORE_THR) ? obj : -1.0f;     // mask0, same -1 fill as reference
  }

  int vcount = 0;
  __syncthreads();

  // ---- 100 serial greedy steps ----
  for (int t = 0; t < MAXOUT; ++t) {
    // local argmax over 16 register scores (ascending element index => first-max tiebreak)
    float v = sc[0]; int li = 0;
#pragma unroll
    for (int j = 1; j < KPT; ++j)
      if (sc[j] > v) { v = sc[j]; li = j; }
    int gi = li * NTHREADS + tid;                 // global element index

    // wave32 butterfly reduce (value, lowest-index tiebreak == jnp.argmax)
#pragma unroll
    for (int off = 16; off >= 1; off >>= 1) {
      float ov = __shfl_xor(v, off, 32);
      int   oi = __shfl_xor(gi, off, 32);
      if (ov > v || (ov == v && oi < gi)) { v = ov; gi = oi; }
    }
    if (lane == 0) { s_redv[wid] = v; s_redi[wid] = gi; }
    __syncthreads();                              // (A)

    if (tid < 32) {
      float v2 = s_redv[tid]; int gi2 = s_redi[tid];
#pragma unroll
      for (int off = 16; off >= 1; off >>= 1) {
        float ov = __shfl_xor(v2, off, 32);
        int   oi = __shfl_xor(gi2, off, 32);
        if (ov > v2 || (ov == v2 && oi < gi2)) { v2 = ov; gi2 = oi; }
      }
      if (tid == 0) { s_bestv = v2; s_besti = gi2; }
    }
    __syncthreads();                              // (B)

    const int   besti = s_besti;
    const float bestv = s_bestv;
    if (tid == (besti & (NTHREADS - 1))) {        // owner publishes winner box
      const int j = besti / NTHREADS;
      s_box[0] = bx1[j]; s_box[1] = by1[j];
      s_box[2] = bx2[j]; s_box[3] = by2[j];
    }
    __syncthreads();                              // (C)

    const bool  valid = (bestv >= 0.0f);
    const float wx1 = s_box[0], wy1 = s_box[1], wx2 = s_box[2], wy2 = s_box[3];

    if (tid == 0 && valid) {                      // stage outputs (invalid slots stay 0)
      s_stage[t * 4 + 0] = fminf(fmaxf(wx1, 0.0f), 1.0f);
      s_stage[t * 4 + 1] = fminf(fmaxf(wy1, 0.0f), 1.0f);
      s_stage[t * 4 + 2] = fminf(fmaxf(wx2, 0.0f), 1.0f);
      s_stage[t * 4 + 3] = fminf(fmaxf(wy2, 0.0f), 1.0f);
      s_stage[400 + t]   = bestv;                 // masked==raw score when valid
      ++vcount;
    }

    if (valid) {                                  // suppress in registers (incl. self)
      const float wa = (wx2 - wx1) * (wy2 - wy1);
#pragma unroll
      for (int j = 0; j < KPT; ++j) {
        const float xx1 = fmaxf(wx1, bx1[j]);
        const float yy1 = fmaxf(wy1, by1[j]);
        const float xx2 = fminf(wx2, bx2[j]);
        const float yy2 = fminf(wy2, by2[j]);
        const float inter = fmaxf(xx2 - xx1, 0.0f) * fmaxf(yy2 - yy1, 0.0f);
        const float iou = inter / fmaxf(wa + barea[j] - inter, 1e-9f);
        if (iou > IOU_THR) sc[j] = -1.0f;
      }
    }
  }

  if (tid == 0) s_stage[600] = (float)vcount;
  __syncthreads();   // DS writes visible before async DMA reads LDS

  // ---- CDNA5 async LDS->global bulk flush of staged outputs ----
  if (tid < 601) {
    float* g;
    if (tid < 400)      g = out + (size_t)b * (MAXOUT * 4) + tid;
    else if (tid < 500) g = out + NS_OFF + (size_t)b * MAXOUT + (tid - 400);
    else if (tid < 600) g = out + CL_OFF + (size_t)b * MAXOUT + (tid - 500);
    else                g = out + VD_OFF + b;
    const uint32_t laddr = (uint32_t)(uintptr_t)(&s_stage[tid]); // flat[31:0] == LDS addr
    const uint64_t gaddr = (uint64_t)(uintptr_t)g;
    asm volatile("global_store_async_from_lds_b32 %0, %1, off"
                 :: "v"(gaddr), "v"(laddr) : "memory");
  }
  asm volatile("s_wait_asynccnt 0" ::: "memory");
}

extern "C" void kernel_launch(void* const* d_in, const int* in_sizes, int n_in,
                              void* d_out, int out_size, void* d_ws, size_t ws_size,
                              hipStream_t stream) {
  (void)in_sizes; (void)n_in; (void)d_ws; (void)ws_size; (void)out_size;
  const float* in = (const float*)d_in[0];
  float* out = (float*)d_out;
  nms_gfx1250_kernel<<<BATCH, NTHREADS, 0, stream>>>(in, out);
}